// GNNs_18193481466441
// MI455X (gfx1250) — compile-verified
//
#include <hip/hip_runtime.h>

// ---------------------------------------------------------------------------
// 3-layer GCN (DGL GraphConv, norm='both') for MI455X / gfx1250.
//
// Cost model: GEMMs total ~11 GFLOP (trivial); the dominant cost is the edge
// aggregation: 3 x 160k edges x 512 feats of gather + f32 atomic scatter-add,
// ~2.5 GB of traffic that fits entirely in the 192 MB L2. The kernel is
// L2/atomic bound, so we keep full fp32 precision and use the fp32 WMMA path
// (v_wmma_f32_16x16x4_f32) for the dense layers, and hardware
// global_atomic_add_f32 (unsafeAtomicAdd) for the scatter.
// ---------------------------------------------------------------------------

#define NNODES   10000
#define NEDGES   160000
#define INFEATS  512
#define NHIDDEN  512
#define NCLASSES 64

typedef __attribute__((ext_vector_type(2))) float v2f;
typedef __attribute__((ext_vector_type(8))) float v8f;

// Native hardware f32 atomic add (global_atomic_add_f32), not a CAS loop.
__device__ __forceinline__ void atomic_add_hw(float* p, float v) {
    unsafeAtomicAdd(p, v);
}

// ---------------------------------------------------------------- utilities
__global__ __launch_bounds__(256) void fill_zero_f32(float* __restrict__ p, int n) {
    int i = blockIdx.x * blockDim.x + threadIdx.x;
    if (i < n) p[i] = 0.0f;
}

__global__ __launch_bounds__(256) void fill_zero_f32x4(float* __restrict__ p, int n4) {
    int i = blockIdx.x * blockDim.x + threadIdx.x;
    if (i < n4) ((float4*)p)[i] = make_float4(0.f, 0.f, 0.f, 0.f);
}

// deg_out[src[e]] += 1 ; deg_in[dst[e]] += 1
__global__ __launch_bounds__(256) void degree_kernel(const int* __restrict__ src,
                                                     const int* __restrict__ dst,
                                                     float* __restrict__ deg_out,
                                                     float* __restrict__ deg_in,
                                                     int nEdges) {
    int e = blockIdx.x * blockDim.x + threadIdx.x;
    if (e >= nEdges) return;
    atomic_add_hw(&deg_out[src[e]], 1.0f);
    atomic_add_hw(&deg_in[dst[e]], 1.0f);
}

// x = clip(x,1,inf)^-0.5  in place (applied to both norm arrays contiguously)
__global__ __launch_bounds__(256) void norm_kernel(float* __restrict__ p, int n) {
    int i = blockIdx.x * blockDim.x + threadIdx.x;
    if (i >= n) return;
    float d = p[i];
    d = fmaxf(d, 1.0f);
    p[i] = rsqrtf(d);
}

// m[dst[e]][f] += h[src[e]][f] * norm_src[src[e]]   (float4 per thread)
__global__ __launch_bounds__(256) void aggregate_kernel(const float* __restrict__ h,
                                                        const int* __restrict__ src,
                                                        const int* __restrict__ dst,
                                                        const float* __restrict__ norm_src,
                                                        float* __restrict__ m,
                                                        int nEdges, int F) {
    const int chunks = F >> 2;                    // float4 chunks per edge row
    int idx = blockIdx.x * blockDim.x + threadIdx.x;
    if (idx >= nEdges * chunks) return;
    int e = idx / chunks;
    int c = idx - e * chunks;
    int s = src[e];
    int d = dst[e];
    float ns = norm_src[s];
    const float4 v = *(const float4*)(h + s * F + c * 4);
    float* mp = m + d * F + c * 4;
    atomic_add_hw(mp + 0, v.x * ns);
    atomic_add_hw(mp + 1, v.y * ns);
    atomic_add_hw(mp + 2, v.z * ns);
    atomic_add_hw(mp + 3, v.w * ns);
}

// ---------------------------------------------------------------- WMMA GEMM
// out[r][c] = act( (A[r][:] * norm_dst[r]) @ W[:,c] + bias[c] )
// Each wave computes a 64(M) x 32(N) tile via 4x2 accumulators of
// v_wmma_f32_16x16x4_f32, K stepped by 4.
//
// fp32 16x16x4 WMMA VGPR layouts (ISA 7.12.2):
//   A 16x4 : lane<16 -> M=lane,     VGPR{0,1} = K{0,1}
//            lane>=16-> M=lane-16,  VGPR{0,1} = K{2,3}
//   B 4x16 : mirrored (lane%16 = N, lane-half selects K pair)
//   C/D    : VGPR v -> M = v + 8*(lane>=16), N = lane%16
__global__ __launch_bounds__(256) void gcn_gemm_wmma(const float* __restrict__ Ain,
                                                     const float* __restrict__ norm_dst,
                                                     const float* __restrict__ W,
                                                     const float* __restrict__ bias,
                                                     float* __restrict__ out,
                                                     int nRows, int K, int N, int doRelu) {
    const int lane = threadIdx.x & 31;
    const int wave = threadIdx.x >> 5;
    const int tilesN = N >> 5;                 // 32-wide column strips
    const int tilesM = (nRows + 63) >> 6;      // 64-tall row strips
    const int w = blockIdx.x * 8 + wave;
    if (w >= tilesM * tilesN) return;          // wave-uniform: EXEC stays all-1s
    const int m0 = (w / tilesN) << 6;
    const int n0 = (w % tilesN) << 5;

    const int lm = lane & 15;                  // N (for B/C) or M (for A) within tile
    const int lh = lane >> 4;                  // lane-half selects K pair / M+8

    // Row indices + fused norm_dst scale; dead rows get scale 0 (clamped addr).
    int rows[4];
    float sc[4];
#pragma unroll
    for (int t = 0; t < 4; ++t) {
        int r = m0 + 16 * t + lm;
        bool valid = r < nRows;
        rows[t] = valid ? r : 0;
        sc[t] = valid ? norm_dst[r] : 0.0f;
    }
    const int cn0 = n0 + lm;

    v8f acc[4][2];
#pragma unroll
    for (int t = 0; t < 4; ++t)
#pragma unroll
        for (int u = 0; u < 2; ++u)
#pragma unroll
            for (int v = 0; v < 8; ++v) acc[t][u][v] = 0.0f;

    for (int k = 0; k < K; k += 4) {
        const int kk = k + (lh << 1);
        v2f a[4], b[2];
#pragma unroll
        for (int t = 0; t < 4; ++t) {
            const float2 av = *(const float2*)(Ain + rows[t] * K + kk);
            a[t].x = av.x * sc[t];
            a[t].y = av.y * sc[t];
        }
#pragma unroll
        for (int u = 0; u < 2; ++u) {
            const float* wp = W + kk * N + cn0 + 16 * u;
            b[u].x = wp[0];
            b[u].y = wp[N];
        }
#pragma unroll
        for (int t = 0; t < 4; ++t)
#pragma unroll
            for (int u = 0; u < 2; ++u)
                acc[t][u] = __builtin_amdgcn_wmma_f32_16x16x4_f32(
                    /*neg_a=*/false, a[t], /*neg_b=*/false, b[u],
                    /*c_mod=*/(short)0, acc[t][u],
                    /*reuse_a=*/false, /*reuse_b=*/false);
    }

    // Epilogue: bias + optional ReLU, row-guarded stores.
#pragma unroll
    for (int t = 0; t < 4; ++t) {
#pragma unroll
        for (int u = 0; u < 2; ++u) {
            const int col = n0 + 16 * u + lm;
            const float bv = bias[col];
#pragma unroll
            for (int v = 0; v < 8; ++v) {
                int row = m0 + 16 * t + lh * 8 + v;
                if (row < nRows) {
                    float val = acc[t][u][v] + bv;
                    if (doRelu) val = fmaxf(val, 0.0f);
                    out[row * N + col] = val;
                }
            }
        }
    }
}

// ---------------------------------------------------------------- launcher
static inline void run_layer(const float* h_in, const int* src, const int* dst,
                             const float* norm_src, const float* norm_dst,
                             const float* W, const float* bias, float* mbuf,
                             float* h_out, int N, int doRelu, hipStream_t stream) {
    const int K = NHIDDEN;  // input feature dim is 512 for every layer here
    // zero accumulation buffer
    {
        int n4 = (NNODES * K) / 4;
        fill_zero_f32x4<<<(n4 + 255) / 256, 256, 0, stream>>>(mbuf, n4);
    }
    // edge scatter-add with fused norm_src scaling
    {
        int work = NEDGES * (K / 4);
        aggregate_kernel<<<(work + 255) / 256, 256, 0, stream>>>(h_in, src, dst, norm_src,
                                                                 mbuf, NEDGES, K);
    }
    // fused norm_dst * GEMM + bias + act
    {
        int tilesM = (NNODES + 63) / 64;
        int tilesN = N / 32;
        int waves = tilesM * tilesN;
        int blocks = (waves + 7) / 8;
        gcn_gemm_wmma<<<blocks, 256, 0, stream>>>(mbuf, norm_dst, W, bias, h_out, NNODES, K,
                                                  N, doRelu);
    }
}

extern "C" void kernel_launch(void* const* d_in, const int* in_sizes, int n_in,
                              void* d_out, int out_size, void* d_ws, size_t ws_size,
                              hipStream_t stream) {
    const float* feat = (const float*)d_in[0];
    const int* src = (const int*)d_in[1];
    const int* dst = (const int*)d_in[2];
    const float* W0 = (const float*)d_in[3];
    const float* b0 = (const float*)d_in[4];
    const float* W1 = (const float*)d_in[5];
    const float* b1 = (const float*)d_in[6];
    const float* W2 = (const float*)d_in[7];
    const float* b2 = (const float*)d_in[8];

    float* out = (float*)d_out;
    float* layer_out = out;                        // [NNODES, NCLASSES]
    float* layer_h = out + NNODES * NCLASSES;      // [NNODES, NHIDDEN]

    float* ws = (float*)d_ws;
    float* norm_src = ws;                          // NNODES
    float* norm_dst = ws + NNODES;                 // NNODES
    float* h0 = ws + 2 * NNODES;                   // NNODES*NHIDDEN
    float* mbuf = h0 + NNODES * NHIDDEN;           // NNODES*NHIDDEN

    // degree norms (norm arrays double as degree accumulators)
    fill_zero_f32<<<(2 * NNODES + 255) / 256, 256, 0, stream>>>(norm_src, 2 * NNODES);
    degree_kernel<<<(NEDGES + 255) / 256, 256, 0, stream>>>(src, dst, norm_src, norm_dst,
                                                            NEDGES);
    norm_kernel<<<(2 * NNODES + 255) / 256, 256, 0, stream>>>(norm_src, 2 * NNODES);

    // layer 0: feat -> h0 (relu)
    run_layer(feat, src, dst, norm_src, norm_dst, W0, b0, mbuf, h0, NHIDDEN, 1, stream);
    // layer 1: h0 -> layer_h (relu)
    run_layer(h0, src, dst, norm_src, norm_dst, W1, b1, mbuf, layer_h, NHIDDEN, 1, stream);
    // layer 2: layer_h -> layer_out (no act)
    run_layer(layer_h, src, dst, norm_src, norm_dst, W2, b2, mbuf, layer_out, NCLASSES, 0,
              stream);
}